// TyCatcher_74655121539668
// MI455X (gfx1250) — compile-verified
//
#include <hip/hip_runtime.h>
#include <hip/hip_bf16.h>
#include <math.h>

// ---------------------------------------------------------------------------
// Spatial transformer:
//   kernel 1: theta = MLP(ty_info) via V_WMMA_F32_16X16X4_F32 (one wave32)
//   kernel 2: bilinear grid-sample of rader_map at crop [100:400,100:400]
// ---------------------------------------------------------------------------

typedef __attribute__((ext_vector_type(2))) float v2f;
typedef __attribute__((ext_vector_type(8))) float v8f;

#define BATCH 16
#define CH    16
#define IMH   512
#define IMW   512
#define NCIN  10
#define NHID  64
#define CROP  300
#define CROP0 100

// --------------------------------------------------------------------------
// Kernel 1: one wave (32 lanes). Two chained f32 WMMAs:
//   h0  (16x64) = ty(16x10) @ W0^T(10x64) + b0     -> 4 N-tiles x 3 K-chunks
//   out (16x6)  = h0(16x64) @ W1^T(64x6)  + b1     -> 1 N-tile  x 16 K-chunks
// then theta = [[1+.1*o0, .1*o1, o2],[.1*o3, 1+.1*o4, o5]] per batch row.
//
// WMMA f32 16x16x4 VGPR layouts (ISA 7.12.2), lane l, half = l>>4, j = l&15:
//   A:   a[v] = A[j][2*half + v]          (j = M row)
//   B:   b[v] = B[2*half + v][j]          (j = N col)
//   C/D: c[v] = D[v + 8*half][j]
// --------------------------------------------------------------------------
__global__ void __launch_bounds__(32)
stn_theta_wmma_kernel(const float* __restrict__ ty,   // 16 x 10
                      const float* __restrict__ W0,   // 64 x 10
                      const float* __restrict__ b0,   // 64
                      const float* __restrict__ W1,   //  6 x 64
                      const float* __restrict__ b1,   //  6
                      float* __restrict__ theta_ws,   // 96 (scratch, read by k2)
                      float* __restrict__ theta_out)  // 96 (d_out tail)
{
    __shared__ float h0[BATCH * NHID];

    const int l    = threadIdx.x;   // 0..31
    const int half = l >> 4;        // 0 or 1
    const int j    = l & 15;        // A-row (M) / B,C-column (N)

    // ---- layer 0: h0 = ty @ W0^T + b0 ------------------------------------
    for (int nt = 0; nt < NHID / 16; ++nt) {
        v8f acc = {};
        for (int kc = 0; kc < 3; ++kc) {            // K = 10 padded to 12
            v2f a, b;
#pragma unroll
            for (int v = 0; v < 2; ++v) {
                const int k = kc * 4 + half * 2 + v;
                a[v] = (k < NCIN) ? ty[j * NCIN + k] : 0.0f;
                // B[k][n] = W0^T[k][nt*16+j] = W0[nt*16+j][k]
                b[v] = (k < NCIN) ? W0[(nt * 16 + j) * NCIN + k] : 0.0f;
            }
            acc = __builtin_amdgcn_wmma_f32_16x16x4_f32(
                false, a, false, b, (short)0, acc, false, false);
        }
#pragma unroll
        for (int v = 0; v < 8; ++v) {
            const int row = v + 8 * half;           // batch row
            h0[row * NHID + nt * 16 + j] = acc[v] + b0[nt * 16 + j];
        }
    }
    __syncthreads();

    // ---- layer 1: out = h0 @ W1^T + b1  (N = 6 padded to 16) -------------
    v8f acc = {};
    for (int kc = 0; kc < NHID / 4; ++kc) {
        v2f a, b;
#pragma unroll
        for (int v = 0; v < 2; ++v) {
            const int k = kc * 4 + half * 2 + v;
            a[v] = h0[j * NHID + k];
            b[v] = (j < 6) ? W1[j * NHID + k] : 0.0f;   // B[k][n] = W1[n][k]
        }
        acc = __builtin_amdgcn_wmma_f32_16x16x4_f32(
            false, a, false, b, (short)0, acc, false, false);
    }

    // ---- theta assembly: lane column j<6 holds out[:, j] for 8 batches ----
    if (j < 6) {
        const float bias = b1[j];
#pragma unroll
        for (int v = 0; v < 8; ++v) {
            const int bidx = v + 8 * half;          // batch index
            const float o  = acc[v] + bias;
            float t;
            if (j == 0 || j == 4)      t = 1.0f + 0.1f * o;  // eye + 0.1*out1
            else if (j == 1 || j == 3) t = 0.1f * o;
            else                       t = o;                 // translation
            theta_ws[bidx * 6 + j]  = t;
            theta_out[bidx * 6 + j] = t;
        }
    }
}

// --------------------------------------------------------------------------
// Kernel 2: one thread per cropped output pixel (b, py, px); coordinates and
// bilinear weights computed once, 16 channels gathered in a loop.
// --------------------------------------------------------------------------
__global__ void __launch_bounds__(256)
stn_sample_kernel(const float* __restrict__ img,    // 16 x 16 x 512 x 512
                  const float* __restrict__ theta,  // 16 x 6
                  float* __restrict__ out)          // 16 x 16 x 300 x 300
{
    const long tid   = (long)blockIdx.x * blockDim.x + threadIdx.x;
    const long total = (long)BATCH * CROP * CROP;
    if (tid >= total) return;

    const int px = (int)(tid % CROP);
    const int py = (int)((tid / CROP) % CROP);
    const int b  = (int)(tid / ((long)CROP * CROP));

    const float* tb = theta + b * 6;
    const float t00 = tb[0], t01 = tb[1], t02 = tb[2];
    const float t10 = tb[3], t11 = tb[4], t12 = tb[5];

    // normalized coords of full-image pixel (px+100, py+100)
    const float X = ((px + CROP0) * 2 + 1) * (1.0f / IMW) - 1.0f;
    const float Y = ((py + CROP0) * 2 + 1) * (1.0f / IMH) - 1.0f;

    const float gx = t00 * X + t01 * Y + t02;
    const float gy = t10 * X + t11 * Y + t12;

    const float ix = ((gx + 1.0f) * IMW - 1.0f) * 0.5f;
    const float iy = ((gy + 1.0f) * IMH - 1.0f) * 0.5f;

    const float x0f = floorf(ix);
    const float y0f = floorf(iy);
    const float x1f = x0f + 1.0f;
    const float y1f = y0f + 1.0f;

    const float wx1 = ix - x0f, wx0 = 1.0f - wx1;
    const float wy1 = iy - y0f, wy0 = 1.0f - wy1;

    // zero-padding masks (float compares, before int conversion)
    const float mx0 = (x0f >= 0.0f && x0f <= (float)(IMW - 1)) ? 1.0f : 0.0f;
    const float mx1 = (x1f >= 0.0f && x1f <= (float)(IMW - 1)) ? 1.0f : 0.0f;
    const float my0 = (y0f >= 0.0f && y0f <= (float)(IMH - 1)) ? 1.0f : 0.0f;
    const float my1 = (y1f >= 0.0f && y1f <= (float)(IMH - 1)) ? 1.0f : 0.0f;

    // clip-in-float then cast (matches jnp.clip(...).astype(int32))
    const int x0 = (int)fminf(fmaxf(x0f, 0.0f), (float)(IMW - 1));
    const int x1 = (int)fminf(fmaxf(x1f, 0.0f), (float)(IMW - 1));
    const int y0 = (int)fminf(fmaxf(y0f, 0.0f), (float)(IMH - 1));
    const int y1 = (int)fminf(fmaxf(y1f, 0.0f), (float)(IMH - 1));

    const float w00 = my0 * mx0 * wy0 * wx0;
    const float w01 = my0 * mx1 * wy0 * wx1;
    const float w10 = my1 * mx0 * wy1 * wx0;
    const float w11 = my1 * mx1 * wy1 * wx1;

    const long o00 = (long)y0 * IMW + x0;
    const long o01 = (long)y0 * IMW + x1;
    const long o10 = (long)y1 * IMW + x0;
    const long o11 = (long)y1 * IMW + x1;

    const float* base = img + (long)b * CH * IMH * IMW;
    float* outp = out + (long)b * CH * CROP * CROP + (long)py * CROP + px;

#pragma unroll 4
    for (int c = 0; c < CH; ++c) {
        const float* pc = base + (long)c * IMH * IMW;
        const float v = w00 * pc[o00] + w01 * pc[o01] +
                        w10 * pc[o10] + w11 * pc[o11];
        outp[(long)c * CROP * CROP] = v;
    }
}

// --------------------------------------------------------------------------
extern "C" void kernel_launch(void* const* d_in, const int* in_sizes, int n_in,
                              void* d_out, int out_size, void* d_ws, size_t ws_size,
                              hipStream_t stream)
{
    (void)in_sizes; (void)n_in; (void)out_size; (void)ws_size;

    const float* ty  = (const float*)d_in[0];   // (16,10)
    const float* img = (const float*)d_in[1];   // (16,16,512,512)
    const float* W0  = (const float*)d_in[2];   // (64,10)
    const float* b0  = (const float*)d_in[3];   // (64,)
    const float* W1  = (const float*)d_in[4];   // (6,64)
    const float* b1  = (const float*)d_in[5];   // (6,)

    float* out       = (float*)d_out;
    float* theta_out = out + (long)BATCH * CH * CROP * CROP;  // +23,040,000
    float* theta_ws  = (float*)d_ws;                          // 96 floats

    // Kernel 1: one wave32, WMMA f32 path for the affine head.
    stn_theta_wmma_kernel<<<1, 32, 0, stream>>>(ty, W0, b0, W1, b1,
                                                theta_ws, theta_out);

    // Kernel 2: memory-bound bilinear gather over the crop.
    const long total = (long)BATCH * CROP * CROP;       // 1,440,000
    const int  block = 256;
    const int  grid  = (int)((total + block - 1) / block);
    stn_sample_kernel<<<grid, block, 0, stream>>>(img, theta_ws, out);
}